// DynamicSignCollaboration_57183194579703
// MI455X (gfx1250) — compile-verified
//
#include <hip/hip_runtime.h>
#include <math.h>

typedef __attribute__((ext_vector_type(2))) float v2f;
typedef __attribute__((ext_vector_type(8))) float v8f;

#define N_NODES 50000
#define DIM 32
#define HSTEP 0.1f

// ---------------------------------------------------------------------------
// Setup kernels
// ---------------------------------------------------------------------------
__global__ void zero_f32_kernel(float* __restrict__ p, int n) {
    int i = blockIdx.x * blockDim.x + threadIdx.x;
    if (i < n) p[i] = 0.0f;
}

__global__ void copy_f32_kernel(const float* __restrict__ in, float* __restrict__ out, int n) {
    int i = blockIdx.x * blockDim.x + threadIdx.x;
    if (i < n) out[i] = in[i];
}

__global__ void degree_kernel(const int* __restrict__ dst, float* __restrict__ deg, int E) {
    int e = blockIdx.x * blockDim.x + threadIdx.x;
    if (e < E) {
        __hip_atomic_fetch_add(&deg[dst[e]], 1.0f, __ATOMIC_RELAXED, __HIP_MEMORY_SCOPE_AGENT);
    }
}

// deg -> dis = rsqrt(deg + 1)  (self loop adds 1)
__global__ void rsqrt_deg_kernel(float* __restrict__ deg, int n) {
    int i = blockIdx.x * blockDim.x + threadIdx.x;
    if (i < n) deg[i] = rsqrtf(deg[i] + 1.0f);
}

__global__ void edge_norm_kernel(const int* __restrict__ src, const int* __restrict__ dst,
                                 const float* __restrict__ dis, float* __restrict__ enorm, int E) {
    int e = blockIdx.x * blockDim.x + threadIdx.x;
    if (e < E) enorm[e] = dis[src[e]] * dis[dst[e]];
}

// ---------------------------------------------------------------------------
// Dense part: xw = x @ W via f32 WMMA 16x16x4; also seeds agg with the
// self-loop term:  agg = (1/deg) * xw + b.
// One wave computes one 16-row tile (16 x 32 output) with 2 col-tiles x 8
// K-chunks = 16 v_wmma_f32_16x16x4_f32 ops.
// ---------------------------------------------------------------------------
__global__ void __launch_bounds__(256)
xw_wmma_kernel(const float* __restrict__ xin,   // [N,32]
               const float* __restrict__ W,     // [32,32] row-major
               const float* __restrict__ b,     // [32]
               const float* __restrict__ dis,   // [N] rsqrt(deg)
               float* __restrict__ xw,          // [N,32] out
               float* __restrict__ agg,         // [N,32] out (self term seeded)
               int n_nodes) {
    int wave = (blockIdx.x * blockDim.x + threadIdx.x) >> 5;
    int lane = threadIdx.x & 31;
    int row0 = wave * 16;
    if (row0 >= n_nodes) return;          // wave-uniform: EXEC stays all-1s for WMMA

    int r  = lane & 15;                   // 0..15
    int kh = lane >> 4;                   // 0 or 1 (which half of the K group)

    const float* xrow = xin + (size_t)(row0 + r) * DIM;

    v8f c0 = {};   // output cols 0..15
    v8f c1 = {};   // output cols 16..31
#pragma unroll
    for (int k0 = 0; k0 < DIM; k0 += 4) {
        // A (16x4 f32): lane<16 holds K=k0,k0+1; lane>=16 holds K=k0+2,k0+3
        v2f a;
        a.x = xrow[k0 + 2 * kh];
        a.y = xrow[k0 + 2 * kh + 1];
        // B (4x16 f32): same K split, N = r (+ col-tile offset)
        v2f b0, b1;
        b0.x = W[(k0 + 2 * kh) * DIM + r];
        b0.y = W[(k0 + 2 * kh + 1) * DIM + r];
        b1.x = W[(k0 + 2 * kh) * DIM + 16 + r];
        b1.y = W[(k0 + 2 * kh + 1) * DIM + 16 + r];
        c0 = __builtin_amdgcn_wmma_f32_16x16x4_f32(false, a, false, b0, (short)0, c0, false, false);
        c1 = __builtin_amdgcn_wmma_f32_16x16x4_f32(false, a, false, b1, (short)0, c1, false, false);
    }

    // D layout: VGPR j -> row (j + 8*kh), col = r (lanes 0-15) / r (lanes 16-31, col-16 local)
#pragma unroll
    for (int j = 0; j < 8; ++j) {
        int row = row0 + j + 8 * kh;
        float dn = dis[row];
        float sn = dn * dn;               // 1/deg (self-loop norm)
        float v0 = c0[j];
        float v1 = c1[j];
        size_t base = (size_t)row * DIM;
        xw[base + r]       = v0;
        xw[base + 16 + r]  = v1;
        agg[base + r]      = sn * v0 + b[r];
        agg[base + 16 + r] = sn * v1 + b[16 + r];
    }
}

// ---------------------------------------------------------------------------
// Sparse part: agg[dst] += edge_norm * xw[src].  8 threads per edge, each
// moves one float4 (4 dims) via one gather load + 4 native f32 atomics.
// Whole hot set (~70MB) is L2-resident on MI455X (192MB L2).
// ---------------------------------------------------------------------------
__global__ void __launch_bounds__(256)
edge_scatter_kernel(const int* __restrict__ src, const int* __restrict__ dst,
                    const float* __restrict__ enorm,
                    const float* __restrict__ xw,
                    float* __restrict__ agg, int E) {
    long tid = (long)blockIdx.x * blockDim.x + threadIdx.x;
    int e = (int)(tid >> 3);
    int q = (int)(tid & 7);
    if (e >= E) return;
    int s = src[e];
    int d = dst[e];
    float w = enorm[e];
    float4 v = ((const float4*)(xw + (size_t)s * DIM))[q];
    float* ad = agg + (size_t)d * DIM + 4 * q;
    __hip_atomic_fetch_add(ad + 0, w * v.x, __ATOMIC_RELAXED, __HIP_MEMORY_SCOPE_AGENT);
    __hip_atomic_fetch_add(ad + 1, w * v.y, __ATOMIC_RELAXED, __HIP_MEMORY_SCOPE_AGENT);
    __hip_atomic_fetch_add(ad + 2, w * v.z, __ATOMIC_RELAXED, __HIP_MEMORY_SCOPE_AGENT);
    __hip_atomic_fetch_add(ad + 3, w * v.w, __ATOMIC_RELAXED, __HIP_MEMORY_SCOPE_AGENT);
}

// ---------------------------------------------------------------------------
// RK4 stage combine: k = relu(agg) * sigmoid(t*wt); update acc and the next
// evaluation point (or final state on stage 3).
// ---------------------------------------------------------------------------
__global__ void __launch_bounds__(256)
rk_combine_kernel(const float* __restrict__ agg,
                  const float* __restrict__ wt,
                  float* __restrict__ x,       // state (lives in d_out)
                  float* __restrict__ xe,      // eval-point buffer
                  float* __restrict__ acc,
                  float t, int stage, int n_elem) {
    int i = blockIdx.x * blockDim.x + threadIdx.x;
    if (i >= n_elem) return;
    int dcol = i & (DIM - 1);
    float g = 1.0f / (1.0f + expf(-t * wt[dcol]));
    float a = agg[i];
    float k = (a > 0.0f ? a : 0.0f) * g;
    if (stage == 0) {
        acc[i] = k;
        xe[i] = x[i] + (HSTEP * 0.5f) * k;
    } else if (stage == 1) {
        acc[i] += 2.0f * k;
        xe[i] = x[i] + (HSTEP * 0.5f) * k;
    } else if (stage == 2) {
        acc[i] += 2.0f * k;
        xe[i] = x[i] + HSTEP * k;
    } else {
        x[i] = x[i] + (HSTEP / 6.0f) * (acc[i] + k);
    }
}

// ---------------------------------------------------------------------------
// Host orchestration
// ---------------------------------------------------------------------------
extern "C" void kernel_launch(void* const* d_in, const int* in_sizes, int n_in,
                              void* d_out, int out_size, void* d_ws, size_t ws_size,
                              hipStream_t stream) {
    const float* x_in  = (const float*)d_in[0];
    const int*   ei[2] = {(const int*)d_in[1], (const int*)d_in[2]};
    const float* Wm[2] = {(const float*)d_in[3], (const float*)d_in[6]};
    const float* bv[2] = {(const float*)d_in[4], (const float*)d_in[7]};
    const float* wt[2] = {(const float*)d_in[5], (const float*)d_in[8]};

    const int N = N_NODES;
    const int E = in_sizes[1] / 2;
    const size_t nf = (size_t)N * DIM;

    float* ws    = (float*)d_ws;
    float* dis   = ws;              // N
    float* enorm = dis + N;         // E
    float* xw    = enorm + E;       // N*DIM
    float* agg   = xw + nf;         // N*DIM
    float* acc   = agg + nf;        // N*DIM
    float* xe    = acc + nf;        // N*DIM

    const int TB = 256;
    int gN  = (N + TB - 1) / TB;
    int gE  = (E + TB - 1) / TB;
    int gNF = (int)((nf + TB - 1) / TB);
    // one wave per 16-row tile
    int waves = (N + 15) / 16;
    int gW = (waves * 32 + TB - 1) / TB;
    long scatter_threads = (long)E * 8;
    int gS = (int)((scatter_threads + TB - 1) / TB);

    for (int sgn = 0; sgn < 2; ++sgn) {
        const int* src = ei[sgn];
        const int* dst = ei[sgn] + E;
        float* xs = (float*)d_out + (size_t)sgn * nf;

        // --- setup: degree -> dis -> edge_norm; init state from input x ---
        zero_f32_kernel<<<gN, TB, 0, stream>>>(dis, N);
        degree_kernel<<<gE, TB, 0, stream>>>(dst, dis, E);
        rsqrt_deg_kernel<<<gN, TB, 0, stream>>>(dis, N);
        edge_norm_kernel<<<gE, TB, 0, stream>>>(src, dst, dis, enorm, E);
        copy_f32_kernel<<<gNF, TB, 0, stream>>>(x_in, xs, (int)nf);

        // --- RK4 integration: 10 steps x 4 stages ---
        for (int step = 0; step < 10; ++step) {
            float t0 = step * HSTEP;
            for (int stage = 0; stage < 4; ++stage) {
                const float* xin = (stage == 0) ? xs : xe;
                float ts = t0;
                if (stage == 1 || stage == 2) ts = t0 + HSTEP * 0.5f;
                else if (stage == 3)          ts = t0 + HSTEP;

                xw_wmma_kernel<<<gW, TB, 0, stream>>>(xin, Wm[sgn], bv[sgn], dis, xw, agg, N);
                edge_scatter_kernel<<<gS, TB, 0, stream>>>(src, dst, enorm, xw, agg, E);
                rk_combine_kernel<<<gNF, TB, 0, stream>>>(agg, wt[sgn], xs, xe, acc,
                                                          ts, stage, (int)nf);
            }
        }
    }
}